// EEGSDE_53128745452302
// MI455X (gfx1250) — compile-verified
//
#include <hip/hip_runtime.h>

// Problem constants (match reference)
#define BDIM   16
#define CDIM   64
#define TDIM   1000
#define BC     (BDIM * CDIM)       // 1024
#define NPATH  64
#define NHALF  32
#define QUADS  (TDIM / 4)          // 250 float4 per time-row

// Native 4-wide float vector (HIP's float4 is a class; the nontemporal
// builtins require a real scalar/vector type).
typedef __attribute__((ext_vector_type(4))) float f4;

// paths = x*(1 - theta*dt) +/- sqrt(dt)*sigma*eps + amp*exp(-0.5*((t-jt)/10)^2)
// theta = 2*|theta_scale|, sigma = 0.5*|sigma_scale|, dt = 0.01
//   k1 = 1 - 0.02*|theta_scale|
//   s  = 0.05*|sigma_scale|            (= sqrt(dt)*sigma)
__global__ __launch_bounds__(256) void eeg_sde_paths_kernel(
    const float* __restrict__ x,            // [B,C,T]
    const float* __restrict__ theta_scale,  // [1]
    const float* __restrict__ sigma_scale,  // [1]
    const float* __restrict__ eps,          // [NHALF,B,C,T]
    const float* __restrict__ jump_rand,    // [NPATH,B,C,1]
    const float* __restrict__ jump_mag,     // [NPATH,B,C,1]
    const int*   __restrict__ jump_times,   // [NPATH,B,C,1]
    float*       __restrict__ out)          // [NPATH,B,C,T]
{
    __shared__ float sx[TDIM];      // staged x row (CDNA5 async copy target)
    __shared__ float s_amp[NPATH];  // masked jump amplitude per path
    __shared__ float s_jt[NPATH];   // jump center time per path

    const int tid = threadIdx.x;
    const int bc  = blockIdx.x;     // flat (b*CDIM + c), 0..1023

    // --- Stage x[b,c,:] into LDS with the gfx1250 async global->LDS copy.
    // Each active lane moves its own 16B quad; tracked by ASYNCcnt.
    if (tid < QUADS) {
        const float* gp = x + (size_t)bc * TDIM + 4 * tid;
        unsigned lds_off = (unsigned)(unsigned long long)(&sx[4 * tid]); // low 32b = LDS offset
        unsigned long long ga = (unsigned long long)gp;
        asm volatile("global_load_async_to_lds_b128 %0, %1, off"
                     :: "v"(lds_off), "v"(ga) : "memory");
    }

    // --- Per-(n,b,c) jump parameters (uniform per path within this block).
    if (tid < NPATH) {
        const int j = tid * BC + bc;
        const float amp = (jump_rand[j] < 0.1f) ? 0.2f * jump_mag[j] : 0.0f;
        s_amp[tid] = amp;
        s_jt[tid]  = (float)jump_times[j];
    }

    asm volatile("s_wait_asynccnt 0" ::: "memory");  // async LDS writes complete
    __syncthreads();

    if (tid >= QUADS) return;

    const float k1 = 1.0f - 0.02f * fabsf(theta_scale[0]); // (1 - theta*dt)
    const float s  = 0.05f * fabsf(sigma_scale[0]);        // sqrt(dt)*sigma

    // Base quad (x read from HBM exactly once, via LDS), fold drift in registers.
    f4 bq = *(const f4*)&sx[4 * tid];
    bq *= k1;

    const float t0 = (float)(4 * tid);
    const size_t tq = (size_t)4 * tid;

    #pragma unroll 4
    for (int nh = 0; nh < NHALF; ++nh) {
        // One eps load feeds BOTH antithetic outputs (n and n+32): halves eps
        // traffic. Single-touch data -> non-temporal load (th:NT), keep L2 clean.
        const f4 e = __builtin_nontemporal_load(
            (const f4*)(eps + ((size_t)nh * BC + bc) * TDIM + tq));
        if (nh + 1 < NHALF)  // pull next 4MB-strided eps row into GL2 (global_prefetch_b8)
            __builtin_prefetch(eps + ((size_t)(nh + 1) * BC + bc) * TDIM + tq, 0, 1);

        const float aA = s_amp[nh],         jtA = s_jt[nh];
        const float aB = s_amp[nh + NHALF], jtB = s_jt[nh + NHALF];

        f4 jA = {0.f, 0.f, 0.f, 0.f};
        f4 jB = {0.f, 0.f, 0.f, 0.f};
        if (aA != 0.0f) {  // uniform across block -> coherent skip (~90% of rows)
            const float u0 = (t0        - jtA) * 0.1f, u1 = (t0 + 1.0f - jtA) * 0.1f;
            const float u2 = (t0 + 2.0f - jtA) * 0.1f, u3 = (t0 + 3.0f - jtA) * 0.1f;
            jA.x = aA * __expf(-0.5f * u0 * u0);
            jA.y = aA * __expf(-0.5f * u1 * u1);
            jA.z = aA * __expf(-0.5f * u2 * u2);
            jA.w = aA * __expf(-0.5f * u3 * u3);
        }
        if (aB != 0.0f) {
            const float u0 = (t0        - jtB) * 0.1f, u1 = (t0 + 1.0f - jtB) * 0.1f;
            const float u2 = (t0 + 2.0f - jtB) * 0.1f, u3 = (t0 + 3.0f - jtB) * 0.1f;
            jB.x = aB * __expf(-0.5f * u0 * u0);
            jB.y = aB * __expf(-0.5f * u1 * u1);
            jB.z = aB * __expf(-0.5f * u2 * u2);
            jB.w = aB * __expf(-0.5f * u3 * u3);
        }

        const f4 se = s * e;
        const f4 oA = bq + se + jA;   // path n   (+eps)
        const f4 oB = bq - se + jB;   // path n+32 (-eps, antithetic)

        // Write-once streaming output -> non-temporal stores (th:TH_STORE_NT).
        __builtin_nontemporal_store(oA, (f4*)(out + ((size_t)nh * BC + bc) * TDIM + tq));
        __builtin_nontemporal_store(oB, (f4*)(out + ((size_t)(nh + NHALF) * BC + bc) * TDIM + tq));
    }
}

extern "C" void kernel_launch(void* const* d_in, const int* in_sizes, int n_in,
                              void* d_out, int out_size, void* d_ws, size_t ws_size,
                              hipStream_t stream) {
    const float* x           = (const float*)d_in[0];
    const float* theta_scale = (const float*)d_in[1];
    const float* sigma_scale = (const float*)d_in[2];
    const float* eps         = (const float*)d_in[3];
    const float* jump_rand   = (const float*)d_in[4];
    const float* jump_mag    = (const float*)d_in[5];
    const int*   jump_times  = (const int*)d_in[6];
    float*       out         = (float*)d_out;

    dim3 grid(BC);    // 1024 blocks: one per (b,c) row family
    dim3 block(256);  // 8 wave32 per block
    eeg_sde_paths_kernel<<<grid, block, 0, stream>>>(
        x, theta_scale, sigma_scale, eps, jump_rand, jump_mag, jump_times, out);
}